// MultiChannelGCNConv_27187142983850
// MI455X (gfx1250) — compile-verified
//
#include <hip/hip_runtime.h>
#include <hip/hip_bf16.h>

// ---------------------------------------------------------------------------
// Multi-channel GCNConv for MI455X (gfx1250, wave32).
//   out[:,c,:] = D^-1/2 (A+I) D^-1/2 (x[:,c,:] @ W[c]) + b[c]
// GEMM phase uses V_WMMA_F32_16X16X4_F32 (f32 in/out, 16x16 tiles, K=4/issue).
// Edge phase is wave-per-edge gather + global_atomic_add_f32 scatter; xs and
// out (51 MB each) fit in the 192 MB L2, so the scatter runs out of L2.
// ---------------------------------------------------------------------------

typedef __attribute__((ext_vector_type(2))) float v2f;
typedef __attribute__((ext_vector_type(8))) float v8f;

#define NN 50000          // N_NODES (divisible by 16)
#define NC 4              // channels
#define FI 64             // F_IN
#define FO 64             // F_OUT
#define NSTRIDE (NC * FI) // 256 floats per node (same for x, xs, out)

// ---------------- degree / normalization --------------------------------

__global__ void gcn_deg_init(float* __restrict__ deg) {
    int i = blockIdx.x * blockDim.x + threadIdx.x;
    if (i < NN) deg[i] = 1.0f;   // self-loop weight
}

__global__ void gcn_deg_accum(const long long* __restrict__ ei,  // [2,E]
                              const float* __restrict__ ew,
                              float* __restrict__ deg, int E) {
    int e = blockIdx.x * blockDim.x + threadIdx.x;
    if (e < E) {
        long long col = ei[(long long)E + e];
        atomicAdd(&deg[col], ew[e]);
    }
}

__global__ void gcn_dinv(float* __restrict__ deg) {
    int i = blockIdx.x * blockDim.x + threadIdx.x;
    if (i < NN) {
        float d = deg[i];
        deg[i] = (d > 0.0f) ? rsqrtf(d) : 0.0f;   // in-place -> dinv
    }
}

// ---------------- per-channel GEMM via WMMA f32 16x16x4 ------------------
// One wave computes one 16(node) x 16(out-feature) tile for one channel.
// xs[n,c,o] = dinv[n] * sum_f x[n,c,f] * W[c,f,o]
//
// A (16x4 f32, MxK): lanes 0-15 hold M=lane, VGPR0/1 = K=0/1;
//                    lanes 16-31 hold M=lane-16, VGPR0/1 = K=2/3.
// B (4x16 f32, KxN): VGPR0: lanes0-15 row K=0, lanes16-31 row K=2;
//                    VGPR1: rows K=1 / K=3. (rows striped across lanes)
// D (16x16 f32):     VGPR r: lanes0-15 -> M=r, N=lane; lanes16-31 -> M=r+8.

__global__ void gcn_gemm_wmma(const float* __restrict__ x,
                              const float* __restrict__ W,
                              const float* __restrict__ dinv,
                              float* __restrict__ xs) {
    const int wave   = (blockIdx.x * blockDim.x + threadIdx.x) >> 5;
    const int lane   = threadIdx.x & 31;
    const int half   = lane >> 4;     // 0: lanes 0-15, 1: lanes 16-31
    const int lid    = lane & 15;

    const int n_ot   = FO / 16;                 // 4 output tiles
    const int per16  = NC * n_ot;               // 16 tiles per 16-node group
    const int ntile  = wave / per16;            // 0..3124 (exact: 50000/16)
    const int rem    = wave % per16;
    const int c      = rem / n_ot;
    const int otile  = rem % n_ot;
    if (ntile >= NN / 16) return;               // grid sized exactly; never divergent

    const int node0 = ntile * 16;
    const int o0    = otile * 16;

    const float* __restrict__ xrow = x + (size_t)(node0 + lid) * NSTRIDE + c * FI;
    const float* __restrict__ Wc   = W + (size_t)c * FI * FO;

    v8f acc = {};
    #pragma unroll
    for (int k0 = 0; k0 < FI; k0 += 4) {
        const int ka = k0 + half * 2;
        v2f a, b;
        a.x = xrow[ka + 0];
        a.y = xrow[ka + 1];
        b.x = Wc[(ka + 0) * FO + o0 + lid];
        b.y = Wc[(ka + 1) * FO + o0 + lid];
        // 8 args: (neg_a, A, neg_b, B, c_mod, C, reuse_a, reuse_b)
        acc = __builtin_amdgcn_wmma_f32_16x16x4_f32(
            false, a, false, b, (short)0, acc, false, false);
    }

    // Store D, folding in the source-side dinv[n] scale.
    #pragma unroll
    for (int r = 0; r < 8; ++r) {
        const int m    = r + half * 8;
        const int node = node0 + m;
        xs[(size_t)node * NSTRIDE + c * FI + o0 + lid] = dinv[node] * acc[r];
    }
}

// ---------------- output init: bias + self-loop term ---------------------
// self loop: norm = dinv[n]*1*dinv[n]; msg = xw*norm = dinv[n]*xs[n]

__global__ void gcn_init_out(const float* __restrict__ bias,
                             const float* __restrict__ dinv,
                             const float* __restrict__ xs,
                             float* __restrict__ out) {
    size_t i = (size_t)blockIdx.x * blockDim.x + threadIdx.x;
    if (i >= (size_t)NN * NSTRIDE) return;
    const int node = (int)(i >> 8);       // /256
    const int co   = (int)(i & 255);      // (c,o) flat index into bias[C,F_OUT]
    out[i] = bias[co] + dinv[node] * xs[i];
}

// ---------------- edge scatter: wave-per-edge ----------------------------
// coef = ew[e] * dinv[col];  out[col,:] += coef * xs[row,:]
// 256 floats / 32 lanes = 8 per lane as two float4 gathers + 8 f32 atomics.

__global__ void gcn_scatter(const long long* __restrict__ ei,  // [2,E]
                            const float* __restrict__ ew,
                            const float* __restrict__ dinv,
                            const float* __restrict__ xs,
                            float* __restrict__ out, int E) {
    const int wave = (blockIdx.x * blockDim.x + threadIdx.x) >> 5;
    const int lane = threadIdx.x & 31;
    if (wave >= E) return;

    const long long row = ei[wave];                   // source
    const long long col = ei[(long long)E + wave];    // target
    const float coef = ew[wave] * dinv[col];

    const float* __restrict__ src = xs  + (size_t)row * NSTRIDE;
    float* __restrict__       dst = out + (size_t)col * NSTRIDE;

    #pragma unroll
    for (int it = 0; it < 2; ++it) {
        const int base = it * 128 + lane * 4;
        const float4 v = *(const float4*)(src + base);
        atomicAdd(dst + base + 0, coef * v.x);
        atomicAdd(dst + base + 1, coef * v.y);
        atomicAdd(dst + base + 2, coef * v.z);
        atomicAdd(dst + base + 3, coef * v.w);
    }
}

// ---------------- launch ------------------------------------------------

extern "C" void kernel_launch(void* const* d_in, const int* in_sizes, int n_in,
                              void* d_out, int out_size, void* d_ws, size_t ws_size,
                              hipStream_t stream) {
    const float*     x  = (const float*)d_in[0];        // [N,C,F_IN]
    const long long* ei = (const long long*)d_in[1];    // [2,E] int64
    const float*     ew = (const float*)d_in[2];        // [E]
    const float*     W  = (const float*)d_in[3];        // [C,F_IN,F_OUT]
    const float*     b  = (const float*)d_in[4];        // [C,F_OUT]
    float*           out = (float*)d_out;               // [N,C,F_OUT]
    const int E = in_sizes[2];                          // 800000

    float* deg = (float*)d_ws;            // NN floats; becomes dinv in-place
    float* xs  = deg + NN;                // NN*256 floats (51.2 MB, fits L2)

    const int T = 256;

    gcn_deg_init <<<(NN + T - 1) / T, T, 0, stream>>>(deg);
    gcn_deg_accum<<<(E  + T - 1) / T, T, 0, stream>>>(ei, ew, deg, E);
    gcn_dinv     <<<(NN + T - 1) / T, T, 0, stream>>>(deg);

    // 50000/16 node tiles * 4 channels * 4 output tiles = 50000 waves, 8/block
    const int gemm_waves  = (NN / 16) * NC * (FO / 16);
    gcn_gemm_wmma<<<gemm_waves / 8, T, 0, stream>>>(x, W, deg, xs);

    gcn_init_out <<<(int)(((size_t)NN * NSTRIDE + T - 1) / T), T, 0, stream>>>(b, deg, xs, out);

    // one wave per edge, 8 waves per block
    gcn_scatter  <<<(E + 7) / 8, T, 0, stream>>>(ei, ew, deg, xs, out, E);
}